// PointMaxPooling_77163382440569
// MI455X (gfx1250) — compile-verified
//
#include <hip/hip_runtime.h>

typedef __attribute__((ext_vector_type(2))) float v2f;
typedef __attribute__((ext_vector_type(8))) float v8f;

#define B_   8
#define N_   4096
#define D_   128
#define KNN  16

// One wave (32 threads) per block; block owns 32 query points of one batch.
__global__ __launch_bounds__(32) void knn_maxpool_wmma(
    const float* __restrict__ xyz, const float* __restrict__ data,
    float* __restrict__ outXyz, float* __restrict__ outLocal)
{
    __shared__ float ldsQ[32 * 4];      // query coords (padded to 4)
    __shared__ float ldsQ2[32];         // query squared norms
    __shared__ float ldsDist[64 * 32];  // distance chunk, [cand][row] (row-contig reads)
    __shared__ int   ldsIdx[32 * KNN];  // per-row neighbor indices

    const int lane = threadIdx.x;                    // 0..31
    const int blk  = blockIdx.x;                     // B*N/32 blocks
    const int b    = blk / (N_ / 32);
    const int tileBase = (blk % (N_ / 32)) * 32;     // first query row of this block
    const size_t batchOff = (size_t)b * N_;
    const bool lo16 = (lane < 16);

    // ---- load this lane's query point; publish coords + norm to LDS ----
    const int myRow = tileBase + lane;               // within-batch point index
    const float* qp = xyz + (batchOff + myRow) * 3;
    const float qx = qp[0], qy = qp[1], qz = qp[2];
    ldsQ[lane * 4 + 0] = qx;
    ldsQ[lane * 4 + 1] = qy;
    ldsQ[lane * 4 + 2] = qz;
    ldsQ2[lane] = qx * qx + qy * qy + qz * qz;

    const int nn   = lane & 15;                // N-index within a 16-wide tile
    const int moff = lo16 ? 0 : 8;             // C/D row offset per lane half

    // ---- A matrices for the two query tiles: rows = [-2qx,-2qy,-2qz, 1] ----
    // f32 16x4 A layout: lanes 0-15 hold K=0,1 ; lanes 16-31 hold K=2,3
    v2f a0, a1;
    {
        const int r0 = nn, r1 = 16 + nn;
        const float x0 = ldsQ[r0*4+0], y0 = ldsQ[r0*4+1], z0 = ldsQ[r0*4+2];
        const float x1 = ldsQ[r1*4+0], y1 = ldsQ[r1*4+1], z1 = ldsQ[r1*4+2];
        a0[0] = lo16 ? -2.f * x0 : -2.f * z0;
        a0[1] = lo16 ? -2.f * y0 : 1.f;
        a1[0] = lo16 ? -2.f * x1 : -2.f * z1;
        a1[1] = lo16 ? -2.f * y1 : 1.f;
    }

    // ---- C accumulators: ||q||^2 broadcast across each row ----
    v8f c0, c1;
    #pragma unroll
    for (int v = 0; v < 8; ++v) {
        c0[v] = ldsQ2[v + moff];
        c1[v] = ldsQ2[16 + v + moff];
    }

    // ---- per-lane sorted top-K (ascending) in registers ----
    float hd[KNN]; int hi[KNN];
    #pragma unroll
    for (int i = 0; i < KNN; ++i) { hd[i] = 3.4e38f; hi[i] = myRow; }

    // ---- scan all candidates in chunks of 64 ----
    for (int cb = 0; cb < N_; cb += 64) {
        #pragma unroll
        for (int ct = 0; ct < 4; ++ct) {
            const int n = cb + ct * 16 + nn;
            const float* cp = xyz + (batchOff + n) * 3;
            const float cx = cp[0], cy = cp[1], cz = cp[2];

            // Unguarded speculative prefetch of next chunk's candidate tile:
            // translation faults are silently dropped (ISA 10.5), and the
            // unconditional form folds into the base reg + immediate offset.
            __builtin_prefetch(cp + 64 * 3, 0, 3);

            // B col n = [cx, cy, cz, ||c||^2]; lanes 0-15 hold K=0,1 ; 16-31 hold K=2,3
            // branchless so EXEC stays all-ones (no saveexec in the hot loop)
            const float c2 = cx * cx + cy * cy + cz * cz;
            v2f bm;
            bm[0] = lo16 ? cx : cz;
            bm[1] = lo16 ? cy : c2;

            // D = full squared-distance tile in one WMMA each
            v8f d0 = __builtin_amdgcn_wmma_f32_16x16x4_f32(
                false, a0, false, bm, (short)0, c0, false, false);
            v8f d1 = __builtin_amdgcn_wmma_f32_16x16x4_f32(
                false, a1, false, bm, (short)0, c1, false, false);

            // store transposed [cand][row]; 16B-aligned -> b128 LDS stores
            const int base = (ct * 16 + nn) * 32 + moff;
            *(float4*)&ldsDist[base +  0] = make_float4(d0[0], d0[1], d0[2], d0[3]);
            *(float4*)&ldsDist[base +  4] = make_float4(d0[4], d0[5], d0[6], d0[7]);
            *(float4*)&ldsDist[base + 16] = make_float4(d1[0], d1[1], d1[2], d1[3]);
            *(float4*)&ldsDist[base + 20] = make_float4(d1[4], d1[5], d1[6], d1[7]);
        }

        // single-wave workgroup: LDS ops are in-order within a wave, no barrier needed.
        // Each lane scans its row of the chunk, maintaining sorted top-16.
        for (int c = 0; c < 64; ++c) {
            const float d = ldsDist[c * 32 + lane];
            const int   n = cb + c;
            if (d >= 1e-8f && n != myRow && d < hd[KNN - 1]) {
                hd[KNN - 1] = d; hi[KNN - 1] = n;
                #pragma unroll
                for (int i = KNN - 1; i > 0; --i) {
                    const bool sw = hd[i] < hd[i - 1];
                    const float td = hd[i]; const int ti = hi[i];
                    hd[i]     = sw ? hd[i - 1] : td;
                    hi[i]     = sw ? hi[i - 1] : ti;
                    hd[i - 1] = sw ? td : hd[i - 1];
                    hi[i - 1] = sw ? ti : hi[i - 1];
                }
            }
        }
    }

    // ---- publish neighbor indices ----
    #pragma unroll
    for (int j = 0; j < KNN; ++j) ldsIdx[lane * KNN + j] = hi[j];

    // ---- cooperative gather + max-pool: 32 lanes x 4 dims (coalesced b128) ----
    for (int r = 0; r < 32; ++r) {
        const int qrow = tileBase + r;
        const float* dptr = data + (batchOff + qrow) * D_ + lane * 4;
        float4 m = *(const float4*)dptr;                    // self feature
        #pragma unroll 4
        for (int j = 0; j < KNN; ++j) {
            const int idx = ldsIdx[r * KNN + j];
            const float4 v = *(const float4*)(data + (batchOff + idx) * D_ + lane * 4);
            m.x = fmaxf(m.x, v.x); m.y = fmaxf(m.y, v.y);
            m.z = fmaxf(m.z, v.z); m.w = fmaxf(m.w, v.w);
        }
        *(float4*)(outLocal + (batchOff + qrow) * D_ + lane * 4) = m;
    }

    // ---- xyz passthrough ----
    float* op = outXyz + (batchOff + myRow) * 3;
    op[0] = qx; op[1] = qy; op[2] = qz;
}

extern "C" void kernel_launch(void* const* d_in, const int* in_sizes, int n_in,
                              void* d_out, int out_size, void* d_ws, size_t ws_size,
                              hipStream_t stream) {
    const float* xyz  = (const float*)d_in[0];   // [8,4096,3]  f32
    const float* data = (const float*)d_in[1];   // [8,4096,128] f32
    float* out = (float*)d_out;                  // [xyz | local_data] concat
    float* outXyz   = out;
    float* outLocal = out + (size_t)B_ * N_ * 3;

    const int blocks = (B_ * N_) / 32;           // 1024 blocks, 1 wave each
    knn_maxpool_wmma<<<blocks, 32, 0, stream>>>(xyz, data, outXyz, outLocal);
}